// NodeToGlobal_14620068675881
// MI455X (gfx1250) — compile-verified
//
#include <hip/hip_runtime.h>
#include <math.h>

// Fused segment-stats + WMMA f32 GEMM for MI455X (gfx1250, wave32).
// One block = 16 consecutive graphs = one 16-row WMMA M-tile.
// h read once (512 MB, the roofline limiter); GEMM uses v_wmma_f32_16x16x4_f32.

#define H_DIM   256
#define G_DIM   256
#define THREADS 512
#define SEGS    16

typedef __attribute__((ext_vector_type(2))) float v2f;
typedef __attribute__((ext_vector_type(8))) float v8f;

// Order-preserving float <-> uint key (inputs are finite normals; no NaNs).
__device__ __forceinline__ unsigned fkey(float f) {
  unsigned b = __float_as_uint(f);
  return (b & 0x80000000u) ? ~b : (b | 0x80000000u);
}
__device__ __forceinline__ float funkey(unsigned k) {
  unsigned b = (k & 0x80000000u) ? (k & 0x7fffffffu) : ~k;
  return __uint_as_float(b);
}

__global__ __launch_bounds__(THREADS)
void fused_stats_wmma_kernel(const float* __restrict__ h,
                             const int*   __restrict__ batch,
                             const float* __restrict__ W,
                             const float* __restrict__ bias,
                             float*       __restrict__ out,
                             int N)
{
  // Transposed hc tile: L[k*16 + m], k in [0,1024) (quadrants: sum|minK|maxK|ssq
  // -> mean|min|max|std), m in [0,16). Exactly 64 KB; conflict-free WMMA A reads.
  __shared__ float L[1024 * SEGS];

  const int tid = threadIdx.x;
  const int g0  = blockIdx.x * SEGS;

  // ---- Phase 0: segment boundaries via binary search (batch is sorted) ----
  if (tid <= SEGS) {
    int x = g0 + tid;
    int lo = 0, hi = N;
    while (lo < hi) {
      int mid = (lo + hi) >> 1;
      if (batch[mid] < x) lo = mid + 1; else hi = mid;
    }
    ((int*)L)[tid] = lo;
  }
  __syncthreads();
  const int   rowStart = ((int*)L)[0];
  const int   rowEnd   = ((int*)L)[SEGS];
  const int   m16      = tid & 15;                 // the graph-row this thread finalizes
  const float cntF     = (float)(((int*)L)[m16 + 1] - ((int*)L)[m16]);
  __syncthreads();

  // ---- Phase 1: init stats LDS ----
  const float MN_INIT = __uint_as_float(fkey(__builtin_inff()));    // +inf key
  const float MX_INIT = __uint_as_float(fkey(-__builtin_inff()));   // -inf key
  for (int i = tid; i < 1024 * SEGS; i += THREADS) {
    int q = i >> 12;                               // i = k*16+m -> q = k>>8
    L[i] = (q == 1) ? MN_INIT : (q == 2) ? MX_INIT : 0.0f;
  }
  __syncthreads();

  // ---- Phase 2: one streaming pass over h (8 rows in parallel x 64 float4 cols) ----
  {
    const int c4 = (tid & 63) * 4;                 // column base
    const int rp = tid >> 6;                       // row phase 0..7
    float s[4], q[4], mn[4], mx[4];
    int curSeg = -1;

    auto flush = [&](int seg) {
#pragma unroll
      for (int j = 0; j < 4; ++j) {
        int col = c4 + j;
        atomicAdd(&L[(col)       * 16 + seg], s[j]);
        atomicMin((unsigned*)&L[(256 + col) * 16 + seg], fkey(mn[j]));
        atomicMax((unsigned*)&L[(512 + col) * 16 + seg], fkey(mx[j]));
        atomicAdd(&L[(768 + col) * 16 + seg], q[j]);
      }
    };

    for (int r = rowStart + rp; r < rowEnd; r += 8) {
      int seg = batch[r] - g0;                     // in [0,16) by sortedness
      if (seg != curSeg) {
        if (curSeg >= 0) flush(curSeg);
        curSeg = seg;
#pragma unroll
        for (int j = 0; j < 4; ++j) {
          s[j] = 0.0f; q[j] = 0.0f;
          mn[j] =  __builtin_inff();
          mx[j] = -__builtin_inff();
        }
      }
      const float4 v = *(const float4*)(h + (size_t)r * H_DIM + c4);
      s[0] += v.x; q[0] += v.x * v.x; mn[0] = fminf(mn[0], v.x); mx[0] = fmaxf(mx[0], v.x);
      s[1] += v.y; q[1] += v.y * v.y; mn[1] = fminf(mn[1], v.y); mx[1] = fmaxf(mx[1], v.y);
      s[2] += v.z; q[2] += v.z * v.z; mn[2] = fminf(mn[2], v.z); mx[2] = fmaxf(mx[2], v.z);
      s[3] += v.w; q[3] += v.w * v.w; mn[3] = fminf(mn[3], v.w); mx[3] = fmaxf(mx[3], v.w);
    }
    if (curSeg >= 0) flush(curSeg);
  }
  __syncthreads();

  // ---- Phase 3: finalize in place (mean | min | max | unbiased std) ----
  // Thread owns graph-row m16; covers 8 distinct d values -> no races, no barrier inside.
#pragma unroll
  for (int it = 0; it < 8; ++it) {
    int idx = it * THREADS + tid;                  // (idx & 15) == m16
    int m = idx & 15;
    int d = idx >> 4;                              // 0..255
    float    sum = L[(d)       * 16 + m];
    unsigned mnK = __float_as_uint(L[(256 + d) * 16 + m]);
    unsigned mxK = __float_as_uint(L[(512 + d) * 16 + m]);
    float    ssq = L[(768 + d) * 16 + m];
    float mean = sum / fmaxf(cntF, 1.0f);
    float var  = (ssq - mean * sum) / fmaxf(cntF - 1.0f, 1.0f);
    float stdv = sqrtf(fmaxf(var, 0.0f));
    float mnv  = (cntF > 0.0f) ? funkey(mnK) : 0.0f;
    float mxv  = (cntF > 0.0f) ? funkey(mxK) : 0.0f;
    L[(d)       * 16 + m] = mean;
    L[(256 + d) * 16 + m] = mnv;
    L[(512 + d) * 16 + m] = mxv;
    L[(768 + d) * 16 + m] = stdv;
  }
  __syncthreads();

  // ---- Phase 4: 16x256 = 16 WMMA tiles; wave w owns N-tile w ----
  {
    const int lane  = tid & 31;
    const int wave  = tid >> 5;                    // 0..15
    const int ml    = lane & 15;
    const bool hiG  = lane >= 16;                  // lanes 16..31 carry K+2,K+3
    const int col   = wave * 16 + ml;              // output column (N)

    v8f acc = {0.f, 0.f, 0.f, 0.f, 0.f, 0.f, 0.f, 0.f};

#pragma unroll 4
    for (int k = 0; k < 1024; k += 4) {
      int kk = k + (hiG ? 2 : 0);
      v2f a, b;
      a.x = L[(kk)     * 16 + ml];                 // A: hcT[kk][m], conflict-free
      a.y = L[(kk + 1) * 16 + ml];
      b.x = W[(size_t)(kk)     * G_DIM + col];     // B: W[kk][col] (L2-resident)
      b.y = W[(size_t)(kk + 1) * G_DIM + col];
      acc = __builtin_amdgcn_wmma_f32_16x16x4_f32(
          /*neg_a=*/false, a, /*neg_b=*/false, b,
          /*c_mod=*/(short)0, acc, /*reuse_a=*/false, /*reuse_b=*/false);
    }

    const float bv = bias[col];
#pragma unroll
    for (int r = 0; r < 8; ++r) {
      int row = g0 + r + (hiG ? 8 : 0);            // C/D: vgpr r -> M=r (lo) / r+8 (hi)
      out[(size_t)row * G_DIM + col] = acc[r] + bv;
    }
  }
}

extern "C" void kernel_launch(void* const* d_in, const int* in_sizes, int n_in,
                              void* d_out, int out_size, void* d_ws, size_t ws_size,
                              hipStream_t stream) {
  const float* h     = (const float*)d_in[0];
  const int*   batch = (const int*)d_in[1];
  const float* W     = (const float*)d_in[2];
  const float* b     = (const float*)d_in[3];
  float*       out   = (float*)d_out;

  const int N          = in_sizes[1];              // number of nodes
  const int num_graphs = out_size / G_DIM;         // 8192

  dim3 grid(num_graphs / SEGS);
  dim3 block(THREADS);
  hipLaunchKernelGGL(fused_stats_wmma_kernel, grid, block, 0, stream,
                     h, batch, W, b, out, N);
}